// SIMPLESampler_51101520887988
// MI455X (gfx1250) — compile-verified
//
#include <hip/hip_runtime.h>
#include <stdint.h>

// ---------------- configuration ----------------
#define P   64          // problems (threads) per block: 2 wave32s
#define KK  9           // k+1, k = 8

// LDS layout: each "slot" is P consecutive floats (slot s, problem p -> smem[s*P+p])
// bank = (s*64 + p) % 64 = p  -> conflict-free across a wave.
#define S_TH   0        // 32 slots : theta[c], c in 0..31
#define S_L1   32       // 16 nodes * 4  (3 vals + scale)     supp 2
#define S_L2   96       //  8 nodes * 6  (5 vals + scale)     supp 4
#define S_L3   144      //  4 nodes * 10 (9 vals + scale)     supp 8
#define S_L4   184      //  2 nodes * 10                      supp 8 (truncated)
#define S_L5   204      //  1 node  * 10 (root)
#define S_CNT  214      // 32 slots : counts during sampling AND output staging (marg/samples)
#define S_TOTAL 246     // 246 * 64 * 4B = 62,976 B static LDS

// Truncated linear-space convolution: c[m] = sum_j a[j]*b[m-j], m < min(SA+SB-1, 9).
template <int SA, int SB>
__device__ __forceinline__ void convT(const float* a, const float* b, float* c) {
  constexpr int SC = (SA + SB - 1 < KK) ? (SA + SB - 1) : KK;
#pragma unroll
  for (int m = 0; m < SC; ++m) {
    float acc = 0.f;
#pragma unroll
    for (int j = 0; j < SA; ++j) {
      if ((m - j) >= 0 && (m - j) < SB) acc = fmaf(a[j], b[m - j], acc);
    }
    c[m] = acc;
  }
}

// Normalize v[0..S) by its max; return log(max).
template <int S>
__device__ __forceinline__ float renorm(float* v) {
  float mx = v[0];
#pragma unroll
  for (int i = 1; i < S; ++i) mx = fmaxf(mx, v[i]);
  mx = fmaxf(mx, 1e-30f);
  float inv = 1.0f / mx;
#pragma unroll
  for (int i = 0; i < S; ++i) v[i] *= inv;
  return __logf(mx);
}

// Deterministic counter hash -> uniform [0,1)
__device__ __forceinline__ float rng01(uint32_t x) {
  x ^= x >> 17; x *= 0xed5ad4bbu;
  x ^= x >> 11; x *= 0xac4c1b51u;
  x ^= x >> 15; x *= 0x31848babu;
  x ^= x >> 14;
  return (float)(x >> 8) * (1.0f / 16777216.0f);
}

// One top-down categorical split level (parents j = NPAR-1..0; in-place expansion safe).
template <int E, int NPAR>
__device__ __forceinline__ void split_level(float* sm, int p, int childBase, int stride,
                                            uint32_t seed) {
  for (int j = NPAR - 1; j >= 0; --j) {
    int m = (int)sm[(S_CNT + j) * P + p];
    int ba = (childBase + (2 * j) * stride) * P + p;
    int bb = (childBase + (2 * j + 1) * stride) * P + p;
    float w[E];
    float tot = 0.f;
#pragma unroll
    for (int c = 0; c < E; ++c) {
      int r = m - c;
      float wc = 0.f;
      if (r >= 0 && r < E) wc = sm[ba + c * P] * sm[bb + r * P];
      w[c] = wc;
      tot += wc;
    }
    float u = rng01(seed + (uint32_t)j * 0x68E31DA4u) * tot;
    int pick = -1;
    float acc = 0.f;
#pragma unroll
    for (int c = 0; c < E; ++c) {
      acc += w[c];
      if (pick < 0 && w[c] > 0.f && u <= acc) pick = c;
    }
    if (pick < 0) pick = (m < E) ? m : (E - 1);
    sm[(S_CNT + 2 * j) * P + p] = (float)pick;
    sm[(S_CNT + 2 * j + 1) * P + p] = (float)(m - pick);
  }
}

__global__ __launch_bounds__(P) void simple_sampler_kernel(const float* __restrict__ scores,
                                                           float* __restrict__ out,
                                                           int TOT) {
  __shared__ float smem[S_TOTAL * P];
  const int t = threadIdx.x;
  const int p = t;

  // ---------------- stage in: async B128 copy of 2048 contiguous floats -> LDS (transposed)
  // 4 consecutive source floats (same node, same choice, e0..e0+3) land contiguously in LDS.
  {
    const float* src = scores + (size_t)blockIdx.x * 2048;
#pragma unroll
    for (int it = 0; it < 8; ++it) {
      int q = it * P + t;            // each q covers 4 floats; coalesced per wave
      int nl = q >> 6;               // node_local 0..7
      int c  = (q >> 1) & 31;        // choice
      int e4 = (q & 1) * 4;          // ensemble base (0 or 4)
      unsigned ldsOff = (unsigned)(uintptr_t)(&smem[(S_TH + c) * P + (nl * 8 + e4)]);
      unsigned voff = (unsigned)q * 16u;
      asm volatile("global_load_async_to_lds_b128 %0, %1, %2"
                   :: "v"(ldsOff), "v"(voff), "s"(src)
                   : "memory");
    }
    asm volatile("s_wait_asynccnt 0x0" ::: "memory");
  }
  __syncthreads();

  // ---------------- up pass: linear-space ESP with per-node normalization ----------------
  for (int i = 0; i < 16; ++i) {   // L1: pair of leaves -> 3 entries
    float thA = smem[(S_TH + 2 * i) * P + p];
    float thB = smem[(S_TH + 2 * i + 1) * P + p];
    float a0 = __expf(fminf(-thA, 0.f)), a1 = __expf(fminf(thA, 0.f));
    float b0 = __expf(fminf(-thB, 0.f)), b1 = __expf(fminf(thB, 0.f));
    float C[3];
    C[0] = a0 * b0;
    C[1] = fmaf(a0, b1, a1 * b0);
    C[2] = a1 * b1;
    float ls = fmaxf(thA, 0.f) + fmaxf(thB, 0.f) + renorm<3>(C);
    int bo = (S_L1 + i * 4) * P + p;
    smem[bo] = C[0]; smem[bo + P] = C[1]; smem[bo + 2 * P] = C[2]; smem[bo + 3 * P] = ls;
  }
  for (int i = 0; i < 8; ++i) {    // L2: 3 (x) 3 -> 5
    float A[3], B[3], C[5];
    int ba = (S_L1 + (2 * i) * 4) * P + p, bb = (S_L1 + (2 * i + 1) * 4) * P + p;
#pragma unroll
    for (int q = 0; q < 3; ++q) { A[q] = smem[ba + q * P]; B[q] = smem[bb + q * P]; }
    convT<3, 3>(A, B, C);
    float ls = smem[ba + 3 * P] + smem[bb + 3 * P] + renorm<5>(C);
    int bo = (S_L2 + i * 6) * P + p;
#pragma unroll
    for (int q = 0; q < 5; ++q) smem[bo + q * P] = C[q];
    smem[bo + 5 * P] = ls;
  }
  for (int i = 0; i < 4; ++i) {    // L3: 5 (x) 5 -> 9
    float A[5], B[5], C[9];
    int ba = (S_L2 + (2 * i) * 6) * P + p, bb = (S_L2 + (2 * i + 1) * 6) * P + p;
#pragma unroll
    for (int q = 0; q < 5; ++q) { A[q] = smem[ba + q * P]; B[q] = smem[bb + q * P]; }
    convT<5, 5>(A, B, C);
    float ls = smem[ba + 5 * P] + smem[bb + 5 * P] + renorm<9>(C);
    int bo = (S_L3 + i * 10) * P + p;
#pragma unroll
    for (int q = 0; q < 9; ++q) smem[bo + q * P] = C[q];
    smem[bo + 9 * P] = ls;
  }
  for (int i = 0; i < 2; ++i) {    // L4: 9 (x) 9 -> 9 (truncated at count 8)
    float A[9], B[9], C[9];
    int ba = (S_L3 + (2 * i) * 10) * P + p, bb = (S_L3 + (2 * i + 1) * 10) * P + p;
#pragma unroll
    for (int q = 0; q < 9; ++q) { A[q] = smem[ba + q * P]; B[q] = smem[bb + q * P]; }
    convT<9, 9>(A, B, C);
    float ls = smem[ba + 9 * P] + smem[bb + 9 * P] + renorm<9>(C);
    int bo = (S_L4 + i * 10) * P + p;
#pragma unroll
    for (int q = 0; q < 9; ++q) smem[bo + q * P] = C[q];
    smem[bo + 9 * P] = ls;
  }
  {                                // L5: root
    float A[9], B[9], C[9];
    int ba = (S_L4 + 0) * P + p, bb = (S_L4 + 10) * P + p;
#pragma unroll
    for (int q = 0; q < 9; ++q) { A[q] = smem[ba + q * P]; B[q] = smem[bb + q * P]; }
    convT<9, 9>(A, B, C);
    float ls = smem[ba + 9 * P] + smem[bb + 9 * P] + renorm<9>(C);
    int bo = S_L5 * P + p;
#pragma unroll
    for (int q = 0; q < 9; ++q) smem[bo + q * P] = C[q];
    smem[bo + 9 * P] = ls;
  }

  const float logZ = __logf(fmaxf(smem[(S_L5 + 8) * P + p], 1e-30f)) + smem[(S_L5 + 9) * P + p];

  // ---------------- down pass: marginals -> staged into S_CNT slots ----------------------
  for (int i = 0; i < 16; ++i) {
    float ext[9];
    float es;
    {  // init with L4 sibling table
      int bt = (S_L4 + (((i >> 3) ^ 1)) * 10) * P + p;
#pragma unroll
      for (int q = 0; q < 9; ++q) ext[q] = smem[bt + q * P];
      es = smem[bt + 9 * P];
    }
    {  // L3 sibling
      float T[9], O[9];
      int bt = (S_L3 + (((i >> 2) ^ 1)) * 10) * P + p;
#pragma unroll
      for (int q = 0; q < 9; ++q) T[q] = smem[bt + q * P];
      convT<9, 9>(ext, T, O);
#pragma unroll
      for (int q = 0; q < 9; ++q) ext[q] = O[q];
      es += smem[bt + 9 * P] + renorm<9>(ext);
    }
    {  // L2 sibling
      float T[5], O[9];
      int bt = (S_L2 + (((i >> 1) ^ 1)) * 6) * P + p;
#pragma unroll
      for (int q = 0; q < 5; ++q) T[q] = smem[bt + q * P];
      convT<9, 5>(ext, T, O);
#pragma unroll
      for (int q = 0; q < 9; ++q) ext[q] = O[q];
      es += smem[bt + 5 * P] + renorm<9>(ext);
    }
    {  // L1 sibling
      float T[3], O[9];
      int bt = (S_L1 + ((i ^ 1)) * 4) * P + p;
#pragma unroll
      for (int q = 0; q < 3; ++q) T[q] = smem[bt + q * P];
      convT<9, 3>(ext, T, O);
#pragma unroll
      for (int q = 0; q < 9; ++q) ext[q] = O[q];
      es += smem[bt + 3 * P] + renorm<9>(ext);
    }
    float thL = smem[(S_TH + 2 * i) * P + p];
    float thR = smem[(S_TH + 2 * i + 1) * P + p];
    float vL0 = __expf(fminf(-thL, 0.f)), vL1 = __expf(fminf(thL, 0.f)), mL = fmaxf(thL, 0.f);
    float vR0 = __expf(fminf(-thR, 0.f)), vR1 = __expf(fminf(thR, 0.f)), mR = fmaxf(thR, 0.f);
    float dotL = fmaf(ext[7], vR0, ext[6] * vR1);
    float dotR = fmaf(ext[7], vL0, ext[6] * vL1);
    smem[(S_CNT + 2 * i) * P + p]     = __expf(thL + es + mR - logZ) * dotL;
    smem[(S_CNT + 2 * i + 1) * P + p] = __expf(thR + es + mL - logZ) * dotR;
  }

  // Coalesced async store of marginals (inverse of the load transpose), B128 per lane.
  __syncthreads();
  {
    float* dst = out + (size_t)2 * TOT + (size_t)blockIdx.x * 2048;
#pragma unroll
    for (int it = 0; it < 8; ++it) {
      int q = it * P + t;
      int nl = q >> 6;
      int c  = (q >> 1) & 31;
      int e4 = (q & 1) * 4;
      unsigned ldsOff = (unsigned)(uintptr_t)(&smem[(S_CNT + c) * P + (nl * 8 + e4)]);
      unsigned voff = (unsigned)q * 16u;
      asm volatile("global_store_async_from_lds_b128 %0, %1, %2"
                   :: "v"(voff), "v"(ldsOff), "s"(dst)
                   : "memory");
    }
    asm volatile("s_wait_asynccnt 0x0" ::: "memory");
  }
  __syncthreads();   // S_CNT slots free for reuse by sampling

  // ---------------- top-down exact sampling (2 samples), staged + async stored -----------
  const int b = blockIdx.x * P + p;
  const uint32_t sb = (uint32_t)b * 0x9E3779B9u + 42u;
  for (int s = 0; s < 2; ++s) {
    uint32_t seedL = sb + (uint32_t)(s + 1) * 0x85EBCA6Bu;
    smem[(S_CNT + 0) * P + p] = 8.0f;  // root count = k
    split_level<9, 1>(smem, p, S_L4, 10, seedL ^ 0x1111u);   // root -> 2
    split_level<9, 2>(smem, p, S_L3, 10, seedL ^ 0x2222u);   // 2 -> 4
    split_level<5, 4>(smem, p, S_L2, 6,  seedL ^ 0x3333u);   // 4 -> 8
    split_level<3, 8>(smem, p, S_L1, 4,  seedL ^ 0x4444u);   // 8 -> 16
    // final split to leaves, written in place into S_CNT (descending j is safe)
    for (int j = 15; j >= 0; --j) {
      int m = (int)smem[(S_CNT + j) * P + p];   // m in {0,1,2}
      float thL = smem[(S_TH + 2 * j) * P + p];
      float thR = smem[(S_TH + 2 * j + 1) * P + p];
      float l0 = __expf(fminf(-thL, 0.f)), l1 = __expf(fminf(thL, 0.f));
      float r0 = __expf(fminf(-thR, 0.f)), r1 = __expf(fminf(thR, 0.f));
      float w0 = (m <= 1) ? l0 * ((m == 0) ? r0 : r1) : 0.f;
      float w1 = (m >= 1) ? l1 * ((m == 1) ? r0 : r1) : 0.f;
      float tot = w0 + w1;
      int cL;
      if (tot > 0.f) {
        float u = rng01(seedL ^ (0x5555u + (uint32_t)j * 0x9E37u)) * tot;
        cL = (u < w0) ? 0 : 1;
      } else {
        cL = (m > 0) ? 1 : 0;
      }
      smem[(S_CNT + 2 * j) * P + p]     = (float)cL;
      smem[(S_CNT + 2 * j + 1) * P + p] = (float)(m - cL);
    }
    __syncthreads();
    {
      float* dst = out + (size_t)s * TOT + (size_t)blockIdx.x * 2048;
#pragma unroll
      for (int it = 0; it < 8; ++it) {
        int q = it * P + t;
        int nl = q >> 6;
        int c  = (q >> 1) & 31;
        int e4 = (q & 1) * 4;
        unsigned ldsOff = (unsigned)(uintptr_t)(&smem[(S_CNT + c) * P + (nl * 8 + e4)]);
        unsigned voff = (unsigned)q * 16u;
        asm volatile("global_store_async_from_lds_b128 %0, %1, %2"
                     :: "v"(voff), "v"(ldsOff), "s"(dst)
                     : "memory");
      }
      asm volatile("s_wait_asynccnt 0x0" ::: "memory");
    }
    __syncthreads();   // safe to reuse S_CNT for next sample
  }
}

extern "C" void kernel_launch(void* const* d_in, const int* in_sizes, int n_in,
                              void* d_out, int out_size, void* d_ws, size_t ws_size,
                              hipStream_t stream) {
  (void)n_in; (void)d_ws; (void)ws_size; (void)out_size;
  const float* scores = (const float*)d_in[0];
  float* out = (float*)d_out;
  const int TOT = in_sizes[0];            // nnodes * 32 * 8 = 2,097,152
  const int problems = TOT / 32;          // 65,536
  dim3 grid(problems / P);
  dim3 block(P);
  hipLaunchKernelGGL(simple_sampler_kernel, grid, block, 0, stream, scores, out, TOT);
}